// LengthRegulator_86517821213176
// MI455X (gfx1250) — compile-verified
//
#include <hip/hip_runtime.h>
#include <hip/hip_bf16.h>
#include <stdint.h>

// Problem shape (fixed by the reference: B=16, T=1024, D=512, max_len=4096)
static constexpr int kB      = 16;
static constexpr int kT      = 1024;
static constexpr int kD      = 512;
static constexpr int kMaxLen = 4096;

typedef float v4f __attribute__((ext_vector_type(4)));

// ---------------------------------------------------------------------------
// Kernel A: per-batch inclusive scan of durations (Hillis-Steele in LDS),
// fused with the index scatter: thread i writes "i" into idx[csum[i-1]..csum[i])
// (clipped to max_len). Also emits per-batch totals and mel_len (int64).
// This removes the per-frame binary search from the bandwidth kernel.
// ---------------------------------------------------------------------------
__global__ __launch_bounds__(kT)
void lr_scan_scatter_kernel(const long long* __restrict__ dur,
                            int* __restrict__ totals,
                            int* __restrict__ idx,
                            long long* __restrict__ mel_len) {
    __shared__ int s[kT];
    const int b   = blockIdx.x;
    const int tid = threadIdx.x;

    s[tid] = (int)dur[(size_t)b * kT + tid];
    __syncthreads();

    #pragma unroll
    for (int off = 1; off < kT; off <<= 1) {
        int add = (tid >= off) ? s[tid - off] : 0;
        __syncthreads();
        s[tid] += add;
        __syncthreads();
    }

    const int incl = s[tid];
    const int excl = (tid == 0) ? 0 : s[tid - 1];

    if (tid == kT - 1) {
        totals[b]  = incl;
        mel_len[b] = (long long)incl;   // reference returns unclipped total
    }

    // Scatter this input position into every output frame it covers.
    int start = excl < kMaxLen ? excl : kMaxLen;
    int end   = incl < kMaxLen ? incl : kMaxLen;
    int* row  = idx + (size_t)b * kMaxLen;
    for (int j = start; j < end; ++j) row[j] = tid;
}

// ---------------------------------------------------------------------------
// Kernel B: one wave32 per output frame (frame index uniform per wave).
// One uniform idx load, then a 2KB row copy through the CDNA5 async
// global<->LDS DMA path (ASYNCcnt-tracked, no VGPR data path). Output stores
// are non-temporal: 128 MiB streamed once, keep L2 for the reused x rows.
// ---------------------------------------------------------------------------
__global__ __launch_bounds__(256)
void lr_expand_kernel(const float* __restrict__ x,
                      const int* __restrict__ totals,
                      const int* __restrict__ idx,
                      float* __restrict__ out) {
    __shared__ float smem[8 * kD];              // 8 waves * 2KB = 16KB

    const int wave  = threadIdx.x >> 5;
    const int lane  = threadIdx.x & 31;
    const int frame = blockIdx.x * 8 + wave;    // uniform per wave
    const int b     = frame >> 12;              // frame / 4096
    const int t     = frame & (kMaxLen - 1);    // frame % 4096

    const int total = totals[b];
    float* dst = out + ((size_t)b * kMaxLen + t) * (size_t)kD;

    if (t >= total) {                           // wave-uniform branch: zero-fill
        v4f z = {0.f, 0.f, 0.f, 0.f};
        #pragma unroll
        for (int k = 0; k < 4; ++k)
            __builtin_nontemporal_store(z, (v4f*)(dst + (k * 32 + lane) * 4));
        return;
    }

    const int i = idx[frame];                   // single uniform L2 hit
    const float* src = x + ((size_t)b * kT + i) * (size_t)kD;

    // LDS byte offset of this wave's private 2KB slice (flat-addr low 32 bits)
    const unsigned lds_base = (unsigned)(uintptr_t)(&smem[wave * kD]);

    // Stage the 2KB input row into LDS with async 128-bit DMA loads.
    #pragma unroll
    for (int k = 0; k < 4; ++k) {
        unsigned lds_addr = lds_base + (unsigned)((k * 32 + lane) * 16);
        const float* g    = src + (k * 32 + lane) * 4;
        asm volatile("global_load_async_to_lds_b128 %0, %1, off"
                     :: "v"(lds_addr), "v"(g) : "memory");
    }
    asm volatile("s_wait_asynccnt 0x0" ::: "memory");

    // Drain LDS to the output row with async non-temporal 128-bit DMA stores.
    #pragma unroll
    for (int k = 0; k < 4; ++k) {
        unsigned lds_addr = lds_base + (unsigned)((k * 32 + lane) * 16);
        float* g          = dst + (k * 32 + lane) * 4;
        asm volatile("global_store_async_from_lds_b128 %0, %1, off th:TH_STORE_NT"
                     :: "v"(g), "v"(lds_addr) : "memory");
    }
    asm volatile("s_wait_asynccnt 0x0" ::: "memory");
}

// ---------------------------------------------------------------------------
// Harness entry point.
//   d_in[0]: x        (16*1024*512) float32
//   d_in[1]: duration (16*1024)     int64
//   d_in[2]: max_len  scalar (assumed 4096 per reference)
//   d_out  : out (16*4096*512 f32) ++ mel_len (16 int64) flat
//   d_ws   : totals[16] (int32) ++ idx[16*4096] (int32)  (~256KB)
// ---------------------------------------------------------------------------
extern "C" void kernel_launch(void* const* d_in, const int* in_sizes, int n_in,
                              void* d_out, int out_size, void* d_ws, size_t ws_size,
                              hipStream_t stream) {
    (void)in_sizes; (void)n_in; (void)out_size; (void)ws_size;

    const float*     x   = (const float*)d_in[0];
    const long long* dur = (const long long*)d_in[1];

    float*     out    = (float*)d_out;
    long long* mel    = (long long*)(out + (size_t)kB * kMaxLen * kD);
    int*       totals = (int*)d_ws;
    int*       idx    = totals + 16;

    lr_scan_scatter_kernel<<<kB, kT, 0, stream>>>(dur, totals, idx, mel);

    const int frames = kB * kMaxLen;            // 65536 frames, 8 per block
    lr_expand_kernel<<<frames / 8, 256, 0, stream>>>(x, totals, idx, out);
}